// BackgroundLoss_45432164057702
// MI455X (gfx1250) — compile-verified
//
#include <hip/hip_runtime.h>

// Reference:  attract = mean over valid pids (pid>0, present) of (1 - segmax(beta))
//             + 0.1 * mean(beta | pid==0)
// Only beta (d_in[1]) and particle_id (d_in[4]) are consumed; x/y/w are unused.

#define NUM_PIDS 50000
#define NUM_PIDS_PAD 50048   // multiple of 64; pad entries stay at sentinel -1
#define NB2 32               // blocks for the WMMA table-reduction pass
#define SB_CONST 0.1f

typedef __attribute__((ext_vector_type(2))) float v2f;
typedef __attribute__((ext_vector_type(8))) float v8f;

// ---------------------------------------------------------------------------
// Kernel 0: (re)initialize workspace each launch (graph-replay safe).
// segmax[p] = -1 sentinel: any hit writes a non-negative float-bit pattern,
// so "present" == (segmax >= 0). Positive f32 bits are order-isomorphic to
// signed int, so atomicMax on int == float max for beta in [0,1).
// ---------------------------------------------------------------------------
__global__ void bg_init_kernel(int* __restrict__ segmax,
                               float* __restrict__ psum,
                               unsigned int* __restrict__ pcnt,
                               int nblk) {
  int t = blockIdx.x * blockDim.x + threadIdx.x;
  if (t < NUM_PIDS_PAD) segmax[t] = -1;
  if (t < nblk) { psum[t] = 0.0f; pcnt[t] = 0u; }
}

// ---------------------------------------------------------------------------
// Kernel 1: streaming scatter pass over all hits (the 16 MB bandwidth pass).
// 16B vector loads of beta/pid; pid>0 -> atomicMax into L2-resident table;
// pid==0 -> local noise accumulation, fixed-order LDS tree reduce per block.
// ---------------------------------------------------------------------------
__device__ __forceinline__ void bg_process_hit(int p, float b,
                                               int* __restrict__ segmax,
                                               float& ns, unsigned& nc) {
  if (p == 0) {
    ns += b;
    nc += 1u;
  } else {
    atomicMax(segmax + p, __float_as_int(b));
  }
}

__global__ void bg_scatter_kernel(const float* __restrict__ beta,
                                  const int* __restrict__ pid,
                                  int n,
                                  int* __restrict__ segmax,
                                  float* __restrict__ psum,
                                  unsigned int* __restrict__ pcnt) {
  __shared__ float s_sum[256];
  __shared__ unsigned s_cnt[256];

  const int tid = threadIdx.x;
  const int t = blockIdx.x * blockDim.x + tid;
  const int base = t * 4;

  float ns = 0.0f;
  unsigned nc = 0u;

  if (base + 4 <= n) {
    const float4 b4 = *reinterpret_cast<const float4*>(beta + base);
    const int4   p4 = *reinterpret_cast<const int4*>(pid + base);
    bg_process_hit(p4.x, b4.x, segmax, ns, nc);
    bg_process_hit(p4.y, b4.y, segmax, ns, nc);
    bg_process_hit(p4.z, b4.z, segmax, ns, nc);
    bg_process_hit(p4.w, b4.w, segmax, ns, nc);
  } else {
    for (int i = base; i < n; ++i) {
      bg_process_hit(pid[i], beta[i], segmax, ns, nc);
    }
  }

  // fixed-order tree reduction -> deterministic
  s_sum[tid] = ns;
  s_cnt[tid] = nc;
  __syncthreads();
  for (int st = 128; st > 0; st >>= 1) {
    if (tid < st) {
      s_sum[tid] += s_sum[tid + st];
      s_cnt[tid] += s_cnt[tid + st];
    }
    __syncthreads();
  }
  if (tid == 0) {
    psum[blockIdx.x] = s_sum[0];
    pcnt[blockIdx.x] = s_cnt[0];
  }
}

// ---------------------------------------------------------------------------
// Kernel 2 (NB2 blocks x 8 wave32s): reduce the padded 50K-entry table with
// V_WMMA_F32_16X16X4_F32 (B = ones => every D element is a row-sum of A;
// per-wave total = sum(lane sums) / 16, an exact power-of-2 scale).
// A-slot mapping uses contiguous pairs per lane (i0 = base+2*lane, i1 = i0+1)
// -> a single global_load_b64 per lane per WMMA. Any bijection of the 64
// chunk indices onto A is valid for a pure sum. Table is padded to a multiple
// of 64 with sentinel -1, so loads need no bounds guards: the m>=0 validity
// mask handles the tail for free.
// Fixed block->chunk mapping + fixed-order tree reduce -> deterministic.
// ---------------------------------------------------------------------------
__global__ void bg_table_reduce_kernel(const int* __restrict__ segmax,
                                       float* __restrict__ asum,
                                       unsigned int* __restrict__ avcnt) {
  const int tid  = threadIdx.x;   // 256 threads
  const int wave = tid >> 5;      // 8 waves per block
  const int lane = tid & 31;
  const int gwave = blockIdx.x * 8 + wave;   // 0 .. NB2*8-1

  v8f acc = {};                   // C/D accumulator (16x16 f32, 8 VGPRs)
  v2f bmat;
  bmat[0] = 1.0f;
  bmat[1] = 1.0f;                 // B = all ones (4x16)

  unsigned vcnt = 0u;

  // Wave-uniform trip count; unconditional in-bounds b64 loads (padded table).
  for (int base = gwave * 64; base < NUM_PIDS_PAD; base += NB2 * 8 * 64) {
    const int i0 = base + 2 * lane;      // even index
    // i1 = i0 + 1 (odd, always > 0)

    const int2 m01 = *reinterpret_cast<const int2*>(segmax + i0);
    const int m0 = m01.x;
    const int m1 = m01.y;

    const bool v0 = (i0 > 0) & (m0 >= 0);   // excludes pid 0 (base=0, lane=0)
    const bool v1 = (m1 >= 0);              // odd index, never pid 0

    v2f amat;
    amat[0] = v0 ? (1.0f - __int_as_float(m0)) : 0.0f;
    amat[1] = v1 ? (1.0f - __int_as_float(m1)) : 0.0f;
    vcnt += (v0 ? 1u : 0u) + (v1 ? 1u : 0u);

    // D = A x B + C : accumulates row-sums of A into every column of C.
    acc = __builtin_amdgcn_wmma_f32_16x16x4_f32(
        /*neg_a=*/false, amat, /*neg_b=*/false, bmat,
        /*c_mod=*/(short)0, acc, /*reuse_a=*/false, /*reuse_b=*/false);
  }

  float lane_sum = acc[0] + acc[1] + acc[2] + acc[3] +
                   acc[4] + acc[5] + acc[6] + acc[7];

  __shared__ float    s_a[256];
  __shared__ unsigned s_v[256];
  s_a[tid] = lane_sum;
  s_v[tid] = vcnt;
  __syncthreads();
  for (int st = 128; st > 0; st >>= 1) {
    if (tid < st) {
      s_a[tid] += s_a[tid + st];
      s_v[tid] += s_v[tid + st];
    }
    __syncthreads();
  }
  if (tid == 0) {
    asum[blockIdx.x]  = s_a[0];   // still scaled by 16 (columns); fixed in combine
    avcnt[blockIdx.x] = s_v[0];
  }
}

// ---------------------------------------------------------------------------
// Kernel 3 (1 block): fixed-order combine of NB2 attract partials and the
// per-block noise partials -> final scalar. Fully deterministic.
// ---------------------------------------------------------------------------
__global__ void bg_combine_kernel(const float* __restrict__ asum,
                                  const unsigned int* __restrict__ avcnt,
                                  const float* __restrict__ psum,
                                  const unsigned int* __restrict__ pcnt,
                                  int nblk,
                                  float* __restrict__ out) {
  const int tid = threadIdx.x;    // 256 threads

  // Strided fixed-order pass over noise partials, then fixed tree reduce.
  float nsum = 0.0f;
  unsigned ncnt = 0u;
  for (int i = tid; i < nblk; i += 256) {
    nsum += psum[i];
    ncnt += pcnt[i];
  }

  __shared__ float    s_n[256];
  __shared__ unsigned s_c[256];
  s_n[tid] = nsum;
  s_c[tid] = ncnt;
  __syncthreads();
  for (int st = 128; st > 0; st >>= 1) {
    if (tid < st) {
      s_n[tid] += s_n[tid + st];
      s_c[tid] += s_c[tid + st];
    }
    __syncthreads();
  }

  if (tid == 0) {
    float a = 0.0f;
    unsigned av = 0u;
    for (int i = 0; i < NB2; ++i) {   // serial, fixed order
      a  += asum[i];
      av += avcnt[i];
    }
    a *= (1.0f / 16.0f);  // 16 identical columns per WMMA accumulator
    const float attract = a / (float)(av > 0u ? av : 1u);
    const float noise   = s_n[0] / (float)(s_c[0] > 0u ? s_c[0] : 1u);
    out[0] = attract + SB_CONST * noise;
  }
}

// ---------------------------------------------------------------------------
// Launch: inputs are (w, beta, x, y, particle_id); only beta & pid are used.
// Workspace: [ int segmax[50048] | float psum[nblk] | u32 pcnt[nblk] | pad |
//              float asum[NB2] | u32 avcnt[NB2] ]
// ---------------------------------------------------------------------------
extern "C" void kernel_launch(void* const* d_in, const int* in_sizes, int n_in,
                              void* d_out, int out_size, void* d_ws, size_t ws_size,
                              hipStream_t stream) {
  const float* beta = (const float*)d_in[1];
  const int*   pid  = (const int*)d_in[4];
  const int n = in_sizes[1];

  const int nvec  = (n + 3) / 4;
  const int nblk1 = (nvec + 255) / 256;

  int* segmax = (int*)d_ws;
  char* p0 = (char*)d_ws + ((size_t)NUM_PIDS_PAD * sizeof(int));     // 200192, 256-aligned
  float* psum = (float*)p0;
  unsigned int* pcnt = (unsigned int*)(psum + nblk1);
  char* p1 = (char*)(pcnt + nblk1);
  p1 = (char*)(((uintptr_t)p1 + 255) & ~(uintptr_t)255);
  float* asum = (float*)p1;
  unsigned int* avcnt = (unsigned int*)(asum + NB2);

  const int init_work = (NUM_PIDS_PAD > nblk1) ? NUM_PIDS_PAD : nblk1;
  bg_init_kernel<<<(init_work + 255) / 256, 256, 0, stream>>>(segmax, psum, pcnt, nblk1);
  bg_scatter_kernel<<<nblk1, 256, 0, stream>>>(beta, pid, n, segmax, psum, pcnt);
  bg_table_reduce_kernel<<<NB2, 256, 0, stream>>>(segmax, asum, avcnt);
  bg_combine_kernel<<<1, 256, 0, stream>>>(asum, avcnt, psum, pcnt, nblk1, (float*)d_out);
}